// SoftPhongShaderShadow_36197984371119
// MI455X (gfx1250) — compile-verified
//
#include <hip/hip_runtime.h>
#include <math.h>

// SoftPhongShaderShadow — fused one-pass shader+blend for MI455X (gfx1250).
// Memory floor ~15 us (360 MB @ 23.3 TB/s). Shade kernel keeps VALU near that
// floor using hardware transcendentals only:
//   v_rsq_f32 for the three normalizes, v_exp_f32 for all exponentials,
//   v_log_f32+v_exp_f32 for spec^shininess, v_rcp_f32 for sigmoid/denominator.
//   Pass 1: pack per-face {3 verts, 3 normals} into 80 B/face table (L2-resident, 8 MB).
//   Pass 2: one thread per pixel, K=4 unrolled, streaming I/O as b128, gathers hit L2,
//           global_prefetch_b8 warms gather lines, 2-D grid -> SGPR image constants.

static constexpr float SIGMA_C = 1e-4f;
static constexpr float GAMMA_C = 1e-4f;
static constexpr float ZNEAR_C = 1.0f;
static constexpr float ZFAR_C  = 100.0f;
static constexpr float EPSB_C  = 1e-10f;
static constexpr float INV_SIGMA = 1.0f / SIGMA_C;
static constexpr float INV_GAMMA = 1.0f / GAMMA_C;
static constexpr float INV_ZRANGE = 1.0f / (ZFAR_C - ZNEAR_C);
static constexpr float LOG2E = 1.44269504088896340736f;

__device__ __forceinline__ float fast_exp(float x) {
    return __builtin_amdgcn_exp2f(x * LOG2E);   // v_exp_f32
}
__device__ __forceinline__ float fast_rsq(float x) {
    return __builtin_amdgcn_rsqf(x);            // v_rsq_f32
}
__device__ __forceinline__ float fast_rcp(float x) {
    return __builtin_amdgcn_rcpf(x);            // v_rcp_f32
}
__device__ __forceinline__ float fast_log2(float x) {
    return __builtin_amdgcn_logf(x);            // v_log_f32
}
__device__ __forceinline__ float fast_exp2(float x) {
    return __builtin_amdgcn_exp2f(x);           // v_exp_f32
}

__global__ __launch_bounds__(256) void pack_faces_kernel(
    const int* __restrict__ faces, const float* __restrict__ verts,
    const float* __restrict__ vnorm, float4* __restrict__ tab, int F)
{
    int f = blockIdx.x * blockDim.x + threadIdx.x;
    if (f >= F) return;
    int i0 = faces[3 * f + 0], i1 = faces[3 * f + 1], i2 = faces[3 * f + 2];
    float v0x = verts[3 * i0 + 0], v0y = verts[3 * i0 + 1], v0z = verts[3 * i0 + 2];
    float v1x = verts[3 * i1 + 0], v1y = verts[3 * i1 + 1], v1z = verts[3 * i1 + 2];
    float v2x = verts[3 * i2 + 0], v2y = verts[3 * i2 + 1], v2z = verts[3 * i2 + 2];
    float n0x = vnorm[3 * i0 + 0], n0y = vnorm[3 * i0 + 1], n0z = vnorm[3 * i0 + 2];
    float n1x = vnorm[3 * i1 + 0], n1y = vnorm[3 * i1 + 1], n1z = vnorm[3 * i1 + 2];
    float n2x = vnorm[3 * i2 + 0], n2y = vnorm[3 * i2 + 1], n2z = vnorm[3 * i2 + 2];
    float4* o = tab + (size_t)f * 5;
    o[0] = make_float4(v0x, v0y, v0z, v1x);
    o[1] = make_float4(v1y, v1z, v2x, v2y);
    o[2] = make_float4(v2z, n0x, n0y, n0z);
    o[3] = make_float4(n1x, n1y, n1z, n2x);
    o[4] = make_float4(n2y, n2z, 0.0f, 0.0f);
}

// GRID2D: grid = (HW/256, N); n = blockIdx.y is an SGPR -> per-image loads scalarize.
template <bool USE_TABLE, bool GRID2D>
__global__ __launch_bounds__(256) void shade_blend_kernel(
    const int*   __restrict__ p2f,
    const float* __restrict__ bary,
    const float* __restrict__ zbuf,
    const float* __restrict__ dists,
    const float* __restrict__ texels,
    const float* __restrict__ vis_maps,
    const float* __restrict__ light_loc,
    const float* __restrict__ light_amb,
    const float* __restrict__ light_dif,
    const float* __restrict__ light_spec,
    const float* __restrict__ cam_center,
    const float* __restrict__ mat_amb,
    const float* __restrict__ mat_dif,
    const float* __restrict__ mat_spec,
    const float* __restrict__ shininess,
    const float4* __restrict__ ftab,
    const int*   __restrict__ faces,
    const float* __restrict__ verts,
    const float* __restrict__ vnorm,
    float* __restrict__ out,
    int NHW, int HW)
{
    int pix, n;
    if (GRID2D) {
        n   = blockIdx.y;
        pix = n * HW + blockIdx.x * blockDim.x + threadIdx.x;
    } else {
        pix = blockIdx.x * blockDim.x + threadIdx.x;
        if (pix >= NHW) return;
        n = pix / HW;
    }

    // Per-image constants (SGPR loads in the GRID2D path)
    float llx = light_loc[3 * n + 0], lly = light_loc[3 * n + 1], llz = light_loc[3 * n + 2];
    float cxx = cam_center[3 * n + 0], cyy = cam_center[3 * n + 1], czz = cam_center[3 * n + 2];
    float amb[3], kd[3], ks[3];
#pragma unroll
    for (int c = 0; c < 3; ++c) {
        amb[c] = mat_amb[3 * n + c] * light_amb[3 * n + c];
        kd[c]  = mat_dif[3 * n + c] * light_dif[3 * n + c];
        ks[c]  = mat_spec[3 * n + c] * light_spec[3 * n + c];
    }
    float shin = shininess[n];

    // Per-pixel streaming loads, all 128-bit
    int4 f4 = ((const int4*)p2f)[pix];
    int fidx[4] = { f4.x, f4.y, f4.z, f4.w };

    if (USE_TABLE) {
        // Warm WGP$ for the 4 gathered face rows (global_prefetch_b8)
#pragma unroll
        for (int k = 0; k < 4; ++k) {
            const float4* fb = ftab + (size_t)(fidx[k] < 0 ? 0 : fidx[k]) * 5;
            __builtin_prefetch((const void*)fb, 0, 3);
        }
    }

    float b[12], t[12];
    {
        float4 q;
        q = ((const float4*)bary)[(size_t)pix * 3 + 0]; b[0] = q.x; b[1] = q.y; b[2]  = q.z; b[3]  = q.w;
        q = ((const float4*)bary)[(size_t)pix * 3 + 1]; b[4] = q.x; b[5] = q.y; b[6]  = q.z; b[7]  = q.w;
        q = ((const float4*)bary)[(size_t)pix * 3 + 2]; b[8] = q.x; b[9] = q.y; b[10] = q.z; b[11] = q.w;
        q = ((const float4*)texels)[(size_t)pix * 3 + 0]; t[0] = q.x; t[1] = q.y; t[2]  = q.z; t[3]  = q.w;
        q = ((const float4*)texels)[(size_t)pix * 3 + 1]; t[4] = q.x; t[5] = q.y; t[6]  = q.z; t[7]  = q.w;
        q = ((const float4*)texels)[(size_t)pix * 3 + 2]; t[8] = q.x; t[9] = q.y; t[10] = q.z; t[11] = q.w;
    }
    float4 zb4 = ((const float4*)zbuf)[pix];
    float4 dd4 = ((const float4*)dists)[pix];
    float zbv[4] = { zb4.x, zb4.y, zb4.z, zb4.w };
    float ddv[4] = { dd4.x, dd4.y, dd4.z, dd4.w };
    float visc[3] = { vis_maps[3 * (size_t)pix + 0], vis_maps[3 * (size_t)pix + 1],
                      vis_maps[3 * (size_t)pix + 2] };

    float colors[4][3], prob[4], zinv[4];
    float one_minus = 1.0f;

#pragma unroll
    for (int k = 0; k < 4; ++k) {
        bool m = fidx[k] >= 0;
        int fi = m ? fidx[k] : 0;

        float v0x, v0y, v0z, v1x, v1y, v1z, v2x, v2y, v2z;
        float n0x, n0y, n0z, n1x, n1y, n1z, n2x, n2y, n2z;
        if (USE_TABLE) {
            const float4* fb = ftab + (size_t)fi * 5;
            float4 q0 = fb[0], q1 = fb[1], q2 = fb[2], q3 = fb[3], q4 = fb[4];
            v0x = q0.x; v0y = q0.y; v0z = q0.z; v1x = q0.w; v1y = q1.x; v1z = q1.y;
            v2x = q1.z; v2y = q1.w; v2z = q2.x; n0x = q2.y; n0y = q2.z; n0z = q2.w;
            n1x = q3.x; n1y = q3.y; n1z = q3.z; n2x = q3.w; n2y = q4.x; n2z = q4.y;
        } else {
            int i0 = faces[3 * fi + 0], i1 = faces[3 * fi + 1], i2 = faces[3 * fi + 2];
            v0x = verts[3 * i0 + 0]; v0y = verts[3 * i0 + 1]; v0z = verts[3 * i0 + 2];
            v1x = verts[3 * i1 + 0]; v1y = verts[3 * i1 + 1]; v1z = verts[3 * i1 + 2];
            v2x = verts[3 * i2 + 0]; v2y = verts[3 * i2 + 1]; v2z = verts[3 * i2 + 2];
            n0x = vnorm[3 * i0 + 0]; n0y = vnorm[3 * i0 + 1]; n0z = vnorm[3 * i0 + 2];
            n1x = vnorm[3 * i1 + 0]; n1y = vnorm[3 * i1 + 1]; n1z = vnorm[3 * i1 + 2];
            n2x = vnorm[3 * i2 + 0]; n2y = vnorm[3 * i2 + 1]; n2z = vnorm[3 * i2 + 2];
        }

        // Mask by zeroing the barycentric weights: 0 * finite = 0, so both
        // interpolated position and normal become exactly 0 (matches reference).
        float b0 = m ? b[3 * k + 0] : 0.0f;
        float b1 = m ? b[3 * k + 1] : 0.0f;
        float b2 = m ? b[3 * k + 2] : 0.0f;
        float pcx = b0 * v0x + b1 * v1x + b2 * v2x;
        float pcy = b0 * v0y + b1 * v1y + b2 * v2y;
        float pcz = b0 * v0z + b1 * v1z + b2 * v2z;
        float pnx = b0 * n0x + b1 * n1x + b2 * n2x;
        float pny = b0 * n0y + b1 * n1y + b2 * n2y;
        float pnz = b0 * n0z + b1 * n1z + b2 * n2z;

        // normalize(pn):  x / max(||x||, 1e-6)  ==  x * rsq(max(dot, 1e-12))
        float dn  = pnx * pnx + pny * pny + pnz * pnz;
        float inl = fast_rsq(fmaxf(dn, 1e-12f));
        float nx = pnx * inl, ny = pny * inl, nz = pnz * inl;

        // direction = normalize(light - pc)
        float dx = llx - pcx, dy = lly - pcy, dz = llz - pcz;
        float dd = dx * dx + dy * dy + dz * dz;
        float idl = fast_rsq(fmaxf(dd, 1e-12f));
        dx *= idl; dy *= idl; dz *= idl;

        float cosA = nx * dx + ny * dy + nz * dz;
        float relc = fmaxf(cosA, 0.0f);

        // reflect = -dir + 2*cos*normal
        float rx = -dx + 2.0f * cosA * nx;
        float ry = -dy + 2.0f * cosA * ny;
        float rz = -dz + 2.0f * cosA * nz;

        // view = normalize(cam - pc)
        float vx = cxx - pcx, vy = cyy - pcy, vz = czz - pcz;
        float vv = vx * vx + vy * vy + vz * vz;
        float ivl = fast_rsq(fmaxf(vv, 1e-12f));
        vx *= ivl; vy *= ivl; vz *= ivl;

        float sc = fmaxf(vx * rx + vy * ry + vz * rz, 0.0f);
        sc = (cosA > 0.0f) ? sc : 0.0f;

        // sc^shin via v_log_f32 + v_exp_f32 (sc >= 0).
        //   sc=0 -> log2=-inf -> sp=0 (matches 0^p, p>0); uniform select for p==0.
        float sp = (sc > 0.0f) ? fast_exp2(shin * fast_log2(sc))
                               : ((shin == 0.0f) ? 1.0f : 0.0f);

#pragma unroll
        for (int c = 0; c < 3; ++c)
            colors[k][c] = (amb[c] + kd[c] * relc * visc[c]) * t[3 * k + c] + ks[c] * sp;

        // prob = sigmoid(-d/SIGMA) * mask = mask / (1 + exp(d/SIGMA))
        prob[k] = m ? fast_rcp(1.0f + fast_exp(ddv[k] * INV_SIGMA)) : 0.0f;
        one_minus *= (1.0f - prob[k]);
        zinv[k] = m ? ((ZFAR_C - zbv[k]) * INV_ZRANGE) : 0.0f;
    }

    float zmax = fmaxf(fmaxf(zinv[0], zinv[1]), fmaxf(zinv[2], zinv[3]));
    zmax = fmaxf(zmax, EPSB_C);
    float delta = fast_exp((EPSB_C - zmax) * INV_GAMMA);
    float denom = delta;
    float wr = delta, wg = delta, wb = delta;  // delta * BG (BG = 1,1,1)
#pragma unroll
    for (int k = 0; k < 4; ++k) {
        float w = prob[k] * fast_exp((zinv[k] - zmax) * INV_GAMMA);
        denom += w;
        wr += w * colors[k][0];
        wg += w * colors[k][1];
        wb += w * colors[k][2];
    }
    float rden = fast_rcp(denom);   // denom >= delta > 0, normal range
    float alpha = 1.0f - one_minus;
    ((float4*)out)[pix] = make_float4(wr * rden, wg * rden, wb * rden, alpha);
}

extern "C" void kernel_launch(void* const* d_in, const int* in_sizes, int n_in,
                              void* d_out, int out_size, void* d_ws, size_t ws_size,
                              hipStream_t stream)
{
    const float* verts  = (const float*)d_in[0];
    const float* vnorm  = (const float*)d_in[1];
    const float* bary   = (const float*)d_in[2];
    const float* zbuf   = (const float*)d_in[3];
    const float* dists  = (const float*)d_in[4];
    const float* texels = (const float*)d_in[5];
    const float* vis    = (const float*)d_in[6];
    const float* lloc   = (const float*)d_in[7];
    const float* lamb   = (const float*)d_in[8];
    const float* ldif   = (const float*)d_in[9];
    const float* lspec  = (const float*)d_in[10];
    const float* cam    = (const float*)d_in[11];
    const float* mamb   = (const float*)d_in[12];
    const float* mdif   = (const float*)d_in[13];
    const float* mspec  = (const float*)d_in[14];
    const float* shin   = (const float*)d_in[15];
    const int*   faces  = (const int*)d_in[16];
    const int*   p2f    = (const int*)d_in[17];

    const int F   = in_sizes[16] / 3;
    const int NHW = in_sizes[17] / 4;
    const int N   = in_sizes[7] / 3;
    const int HW  = NHW / N;

    const int TB = 256;
    const bool use_table = ws_size >= (size_t)F * 5 * sizeof(float4);
    const bool grid2d = (HW % TB == 0) && (N * HW == NHW);

    if (use_table) {
        pack_faces_kernel<<<(F + TB - 1) / TB, TB, 0, stream>>>(
            faces, verts, vnorm, (float4*)d_ws, F);
    }

    if (grid2d) {
        dim3 grid(HW / TB, N);
        if (use_table) {
            shade_blend_kernel<true, true><<<grid, TB, 0, stream>>>(
                p2f, bary, zbuf, dists, texels, vis,
                lloc, lamb, ldif, lspec, cam, mamb, mdif, mspec, shin,
                (const float4*)d_ws, nullptr, nullptr, nullptr,
                (float*)d_out, NHW, HW);
        } else {
            shade_blend_kernel<false, true><<<grid, TB, 0, stream>>>(
                p2f, bary, zbuf, dists, texels, vis,
                lloc, lamb, ldif, lspec, cam, mamb, mdif, mspec, shin,
                nullptr, faces, verts, vnorm,
                (float*)d_out, NHW, HW);
        }
    } else {
        dim3 grid((NHW + TB - 1) / TB);
        if (use_table) {
            shade_blend_kernel<true, false><<<grid, TB, 0, stream>>>(
                p2f, bary, zbuf, dists, texels, vis,
                lloc, lamb, ldif, lspec, cam, mamb, mdif, mspec, shin,
                (const float4*)d_ws, nullptr, nullptr, nullptr,
                (float*)d_out, NHW, HW);
        } else {
            shade_blend_kernel<false, false><<<grid, TB, 0, stream>>>(
                p2f, bary, zbuf, dists, texels, vis,
                lloc, lamb, ldif, lspec, cam, mamb, mdif, mspec, shin,
                nullptr, faces, verts, vnorm,
                (float*)d_out, NHW, HW);
        }
    }
}